// PatchGenerator_56461640073282
// MI455X (gfx1250) — compile-verified
//
#include <hip/hip_runtime.h>
#include <cstdint>

// Problem geometry from setup_inputs() (compile-time constants).
#define IMG_H     1024
#define IMG_W     1024
#define N_BATCH   8
#define N_CHAN    16
#define OUT_SZ    80       // size_y == size_x
#define PAD       32       // pad_x == pad_y
#define TPB       256      // 8 wave32 per WG
#define CHUNK_ROWS 16      // input-window rows staged per chunk
#define MAX_PITCH  (OUT_SZ * 4)   // widest window row (L=4): 320 floats

// LDS byte offset of a generic pointer to __shared__ data: the LDS aperture
// lives in the high 32 bits of the flat address, low 32 bits are the offset.
__device__ __forceinline__ unsigned lds_byte_off(const void* p) {
  return (unsigned)(uintptr_t)p;
}

// CDNA5 async global->LDS copy, 4 bytes per lane, tracked by ASYNCcnt.
// Non-temporal: each input element is consumed exactly once (staged to LDS),
// so don't let the 186 MB stream churn L2.
__device__ __forceinline__ void async_copy_b32(unsigned lds_off, unsigned long long gaddr) {
  asm volatile("global_load_async_to_lds_b32 %0, %1, off th:TH_LOAD_NT"
               :: "v"(lds_off), "v"(gaddr)
               : "memory");
}

template<int L>
__device__ __forceinline__ void run_patch(const float* __restrict__ img,
                                          float* __restrict__ outp,
                                          int R0, int C0,
                                          float* buf, int tid) {
  constexpr int WIN    = OUT_SZ * L;          // window width (floats per row)
  constexpr int PITCH  = WIN;                 // LDS row pitch (floats)
  constexpr int TOT    = CHUNK_ROWS * WIN;    // elems per chunk
  constexpr int OPS    = TOT / TPB;           // async ops per thread per chunk = 5L
  constexpr int YB     = CHUNK_ROWS / L;      // output rows per chunk
  constexpr int NCHUNK = (OUT_SZ * L) / CHUNK_ROWS;   // 5L chunks
  constexpr int BUF_FLOATS = CHUNK_ROWS * PITCH;

  const unsigned buf_base = lds_byte_off(buf);

  // Issue all async loads for one chunk into LDS buffer `which`.
  auto issue = [&](int ch, int which) {
    const unsigned base = buf_base + (unsigned)which * (BUF_FLOATS * 4u);
    const int row_origin = R0 + ch * CHUNK_ROWS;
#pragma unroll
    for (int it = 0; it < OPS; ++it) {
      const int i  = tid + it * TPB;
      const int r  = i / WIN;                 // compile-time WIN -> mul/shift, CSE'd
      const int cc = i - r * WIN;
      int grow = row_origin + r;              // replicate-pad clamp (rows)
      grow = grow < 0 ? 0 : (grow > IMG_H - 1 ? IMG_H - 1 : grow);
      int gcol = C0 + cc;                     // replicate-pad clamp (cols)
      gcol = gcol < 0 ? 0 : (gcol > IMG_W - 1 ? IMG_W - 1 : gcol);
      const unsigned long long ga =
          (unsigned long long)(uintptr_t)(img + (size_t)grow * IMG_W + gcol);
      const unsigned la = base + (unsigned)(r * PITCH + cc) * 4u;
      async_copy_b32(la, ga);
    }
  };

  issue(0, 0);

  constexpr float inv = 1.0f / (float)(L * L);

  for (int ch = 0; ch < NCHUNK; ++ch) {
    if (ch + 1 < NCHUNK) {
      issue(ch + 1, (ch + 1) & 1);
      // Async loads complete in order: <= OPS outstanding means chunk `ch` landed.
      asm volatile("s_wait_asynccnt %0" :: "i"(OPS) : "memory");
    } else {
      asm volatile("s_wait_asynccnt 0" ::: "memory");
    }
    __syncthreads();   // all waves' chunk-`ch` data visible in LDS

    const float* b = buf + (ch & 1) * BUF_FLOATS;
    for (int t = tid; t < YB * OUT_SZ; t += TPB) {
      const int yl = t / OUT_SZ;
      const int x  = t - yl * OUT_SZ;
      float acc = 0.0f;
#pragma unroll
      for (int a = 0; a < L; ++a) {
        const float* rp = b + (yl * L + a) * PITCH + x * L;
        if constexpr (L == 4) {
          const float4 v = *(const float4*)rp;   // ds_load_b128
          acc += (v.x + v.y) + (v.z + v.w);
        } else if constexpr (L == 2) {
          const float2 v = *(const float2*)rp;   // ds_load_b64
          acc += v.x + v.y;
        } else {
          acc += *rp;
        }
      }
      const int yg = ch * YB + yl;
      // Output written once, never re-read: non-temporal store.
      __builtin_nontemporal_store(acc * inv, outp + (size_t)yg * OUT_SZ + x);
    }
    __syncthreads();   // buffer (ch&1) free for reuse at chunk ch+2
  }
}

__global__ __launch_bounds__(TPB)
void PatchGenerator_56461640073282_kernel(const float* __restrict__ in_tensor,
                                          const int* __restrict__ sampled_top,
                                          const int* __restrict__ sampled_left,
                                          const int* __restrict__ sampled_level_idx,
                                          float* __restrict__ out) {
  // Double-buffered staging: 2 x 16 rows x 320 floats = 40 KB LDS
  // -> 8 concurrent WGs per 320 KB WGP.
  __shared__ __align__(16) float buf[2 * CHUNK_ROWS * MAX_PITCH];

  const int wg = blockIdx.x;
  const int k  = wg / N_CHAN;
  const int c  = wg - k * N_CHAN;
  const int b  = k % N_BATCH;

  const int L  = 1 << sampled_level_idx[k];     // COARSE_LEVELS = {1,2,4}
  const int R0 = sampled_top[k]  - PAD;          // padded -> unpadded origin
  const int C0 = sampled_left[k] - PAD;

  const float* img  = in_tensor + ((size_t)b * N_CHAN + c) * ((size_t)IMG_H * IMG_W);
  float*       outp = out       + ((size_t)k * N_CHAN + c) * ((size_t)OUT_SZ * OUT_SZ);
  const int tid = threadIdx.x;

  if (L == 1)      run_patch<1>(img, outp, R0, C0, buf, tid);
  else if (L == 2) run_patch<2>(img, outp, R0, C0, buf, tid);
  else             run_patch<4>(img, outp, R0, C0, buf, tid);
}

extern "C" void kernel_launch(void* const* d_in, const int* in_sizes, int n_in,
                              void* d_out, int out_size, void* d_ws, size_t ws_size,
                              hipStream_t stream) {
  (void)n_in; (void)out_size; (void)d_ws; (void)ws_size;
  const float* in_tensor   = (const float*)d_in[0];
  const int*   top         = (const int*)d_in[1];
  const int*   left        = (const int*)d_in[2];
  const int*   level_idx   = (const int*)d_in[3];
  float*       out         = (float*)d_out;

  const int K = in_sizes[1];          // number of patches (64)
  const int grid = K * N_CHAN;        // one WG per (patch, channel)

  PatchGenerator_56461640073282_kernel<<<grid, TPB, 0, stream>>>(
      in_tensor, top, left, level_idx, out);
}